// GNN_Cluster_31284541784154
// MI455X (gfx1250) — compile-verified
//
#include <hip/hip_runtime.h>
#include <hip/hip_bf16.h>
#include <cstddef>
#include <cstdint>

// ---------------------------------------------------------------------------
// GNN_Cluster for MI455X (gfx1250, wave32, WMMA, async-LDS).
//   1) prep:   Y = X*diag(w), s[i] = sum_f w_f x_if^2
//   2) dist:   G = sigmoid(s_i + s_j - 2*(Y X^T) + b) via V_WMMA_F32_16X16X4_F32
//   3) bce:    deterministic two-stage reduction
//   4) eigh:   cyclic Jacobi (tournament pairing), 6 sweeps; G+V L2-resident
//   5) sort:   bitonic sort of diag(G) ascending; C = V[:,perm]*sqrt(clamp)
//   6) GEMMs:  x_new = C@X ; T = A@C ; A_new = C@T
//              double-buffered GLOBAL_LOAD_ASYNC_TO_LDS_B128 staging +
//              s_wait_asynccnt pipeline + f32 WMMA
// ---------------------------------------------------------------------------

typedef __attribute__((ext_vector_type(2))) float v2f;
typedef __attribute__((ext_vector_type(8))) float v8f;

#define NN    2048
#define NF    64
#define NE    65536

// ---------------- prep: Y = X*diag(w), s[i] = sum w_f x^2 -------------------
__global__ __launch_bounds__(256)
void gnncl_prep(const float* __restrict__ X, const float* __restrict__ W,
                float* __restrict__ Y, float* __restrict__ s) {
  const int lane = threadIdx.x & 31;
  const int row  = blockIdx.x * 8 + (threadIdx.x >> 5);
  float p = 0.0f;
  #pragma unroll
  for (int f = lane; f < NF; f += 32) {
    float xv = X[(size_t)row * NF + f];
    float wv = W[f];
    Y[(size_t)row * NF + f] = xv * wv;
    p += wv * xv * xv;
  }
  #pragma unroll
  for (int off = 16; off > 0; off >>= 1) p += __shfl_xor(p, off, 32);
  if (lane == 0) s[row] = p;
}

// ---------------- dist matrix + sigmoid + BCE partials ----------------------
// grid (16,128), block 256 (8 waves); wave -> one 16x16 tile of G. K=64 only,
// operands are L2/L0-resident (X is 512KB) -> direct global fragment loads.
__global__ __launch_bounds__(256)
void gnncl_dist(const float* __restrict__ Y, const float* __restrict__ X,
                const float* __restrict__ s, const float* __restrict__ bias,
                const float* __restrict__ Gtrue,
                float* __restrict__ G, float* __restrict__ Gb,
                float* __restrict__ partial) {
  const int lane = threadIdx.x & 31;
  const int wave = threadIdx.x >> 5;
  const int r = lane & 15, h = lane >> 4;
  const int i0 = blockIdx.y * 16;
  const int j0 = (blockIdx.x * 8 + wave) * 16;

  v8f acc = {};
  for (int k0 = 0; k0 < NF; k0 += 4) {
    v2f a, b;
    const float* ap = Y + (size_t)(i0 + r) * NF + k0 + 2 * h;
    a.x = ap[0]; a.y = ap[1];
    const float* bp = X + (size_t)(j0 + r) * NF + k0 + 2 * h;  // B = X^T
    b.x = bp[0]; b.y = bp[1];
    acc = __builtin_amdgcn_wmma_f32_16x16x4_f32(false, a, false, b,
                                                (short)0, acc, false, false);
  }

  const float bb = bias[0];
  const float sj = s[j0 + r];
  float bsum = 0.0f;
  #pragma unroll
  for (int v = 0; v < 8; ++v) {
    const int i = i0 + v + 8 * h;
    const int j = j0 + r;
    const float val = s[i] + sj - 2.0f * acc[v] + bb;
    const float g = 1.0f / (1.0f + expf(-val));
    const size_t off = (size_t)i * NN + j;
    G[off]  = g;
    Gb[off] = g;
    const float gc = fminf(fmaxf(g, 1e-7f), 1.0f - 1e-7f);
    const float t = Gtrue[off];
    bsum -= t * logf(gc) + (1.0f - t) * logf(1.0f - gc);
  }

  __shared__ float red[256];
  red[threadIdx.x] = bsum;
  __syncthreads();
  for (int st = 128; st > 0; st >>= 1) {
    if (threadIdx.x < st) red[threadIdx.x] += red[threadIdx.x + st];
    __syncthreads();
  }
  if (threadIdx.x == 0) partial[blockIdx.y * gridDim.x + blockIdx.x] = red[0];
}

__global__ __launch_bounds__(1024)
void gnncl_bce_final(const float* __restrict__ partial, float* __restrict__ loss) {
  __shared__ float red[1024];
  red[threadIdx.x] = partial[threadIdx.x] + partial[threadIdx.x + 1024];
  __syncthreads();
  for (int st = 512; st > 0; st >>= 1) {
    if (threadIdx.x < st) red[threadIdx.x] += red[threadIdx.x + st];
    __syncthreads();
  }
  if (threadIdx.x == 0) loss[0] = red[0] * (1.0f / ((float)NN * (float)NN));
}

// ---------------- Jacobi eigensolver (cyclic, tournament pairing) -----------
__device__ __forceinline__ void gnncl_pair(int r, int k, int& p, int& q) {
  const int m = NN - 1;  // 2047
  if (k == 0) { p = m; q = r % m; }
  else { p = (r + k) % m; q = (r + m - k) % m; }
}

__global__ __launch_bounds__(1024)
void gnncl_jrot(const float* __restrict__ G, float2* __restrict__ rot, int r) {
  const int k = threadIdx.x;                 // 1024 pairs, single block
  int p, q; gnncl_pair(r, k, p, q);
  const float app = G[(size_t)p * NN + p];
  const float aqq = G[(size_t)q * NN + q];
  const float apq = G[(size_t)p * NN + q];
  float c = 1.0f, sv = 0.0f;
  if (fabsf(apq) > 1e-12f * fmaxf(fabsf(app) + fabsf(aqq), 1e-30f)) {
    const float tau = (aqq - app) / (2.0f * apq);
    const float t = ((tau >= 0.0f) ? 1.0f : -1.0f) /
                    (fabsf(tau) + sqrtf(1.0f + tau * tau));
    c = 1.0f / sqrtf(1.0f + t * t);
    sv = t * c;
  }
  rot[k] = make_float2(c, sv);
}

// columns of G and V : M <- M*J   (pairs disjoint -> race-free)
__global__ __launch_bounds__(256)
void gnncl_jcols(float* __restrict__ G, float* __restrict__ V,
                 const float2* __restrict__ rot, int r) {
  const int idx = blockIdx.x * 256 + threadIdx.x;  // 2048*1024 threads
  const int k = idx & 1023;
  const int row = idx >> 10;
  int p, q; gnncl_pair(r, k, p, q);
  const float2 cs = rot[k];
  const size_t ip = (size_t)row * NN + p, iq = (size_t)row * NN + q;
  const float gp = G[ip], gq = G[iq];
  G[ip] = cs.x * gp - cs.y * gq;
  G[iq] = cs.y * gp + cs.x * gq;
  const float vp = V[ip], vq = V[iq];
  V[ip] = cs.x * vp - cs.y * vq;
  V[iq] = cs.y * vp + cs.x * vq;
}

// rows of G : M <- J^T * M  (coalesced along columns)
__global__ __launch_bounds__(256)
void gnncl_jrows(float* __restrict__ G, const float2* __restrict__ rot, int r) {
  const int idx = blockIdx.x * 256 + threadIdx.x;
  const int col = idx & (NN - 1);
  const int k = idx >> 11;
  int p, q; gnncl_pair(r, k, p, q);
  const float2 cs = rot[k];
  const size_t pj = (size_t)p * NN + col, qj = (size_t)q * NN + col;
  const float gp = G[pj], gq = G[qj];
  G[pj] = cs.x * gp - cs.y * gq;
  G[qj] = cs.y * gp + cs.x * gq;
}

__global__ __launch_bounds__(256)
void gnncl_init_identity(float* __restrict__ V) {
  const size_t idx = (size_t)blockIdx.x * 256 + threadIdx.x;
  const int i = (int)(idx >> 11), j = (int)(idx & (NN - 1));
  V[idx] = (i == j) ? 1.0f : 0.0f;
}

// ---------------- eigenvalue sort (bitonic, LDS) + C construction -----------
__global__ __launch_bounds__(1024)
void gnncl_eigsort(const float* __restrict__ G, int* __restrict__ perm,
                   float* __restrict__ sq) {
  __shared__ float key[NN];
  __shared__ int   idx[NN];
  for (int e = threadIdx.x; e < NN; e += 1024) {
    key[e] = G[(size_t)e * NN + e];
    idx[e] = e;
  }
  __syncthreads();
  for (int kk = 2; kk <= NN; kk <<= 1) {
    for (int j = kk >> 1; j > 0; j >>= 1) {
      for (int e = threadIdx.x; e < NN; e += 1024) {
        const int partner = e ^ j;
        if (partner > e) {
          const bool up = ((e & kk) == 0);  // ascending
          const float ka = key[e], kb = key[partner];
          const bool sw = up ? (ka > kb) : (ka < kb);
          if (sw) {
            key[e] = kb; key[partner] = ka;
            const int ti = idx[e]; idx[e] = idx[partner]; idx[partner] = ti;
          }
        }
      }
      __syncthreads();
    }
  }
  for (int e = threadIdx.x; e < NN; e += 1024) {
    perm[e] = idx[e];
    sq[e] = sqrtf(fmaxf(key[e], 1e-6f));  // LAMBDA_CLAMP_MIN
  }
}

__global__ __launch_bounds__(256)
void gnncl_make_C(const float* __restrict__ V, const int* __restrict__ perm,
                  const float* __restrict__ sq, float* __restrict__ C) {
  const size_t idx = (size_t)blockIdx.x * 256 + threadIdx.x;
  const int i = (int)(idx >> 11), j = (int)(idx & (NN - 1));
  C[idx] = V[(size_t)i * NN + perm[j]] * sq[j];
}

// ---------------- adjacency scatter ----------------------------------------
__global__ __launch_bounds__(256)
void gnncl_zero(float* __restrict__ p) {
  p[(size_t)blockIdx.x * 256 + threadIdx.x] = 0.0f;
}

__global__ __launch_bounds__(256)
void gnncl_scatter(const int* __restrict__ ei, const float* __restrict__ ea,
                   float* __restrict__ A) {
  const int e = blockIdx.x * 256 + threadIdx.x;  // NE threads
  const int i = ei[e];
  const int j = ei[NE + e];
  A[(size_t)i * NN + j] = ea[e];
}

// ---------------- f32 WMMA GEMM with async-LDS double buffering -------------
// D = A(MxK)@B(KxN). Block = 128 thr (4 waves, 2x2), block tile 64x64,
// K-stage 32. Stages fetched with GLOBAL_LOAD_ASYNC_TO_LDS_B128 (ASYNCcnt),
// one stage kept in flight (s_wait_asynccnt 8), consumed after barrier.
// LDS strides: A rows 36 dwords (16B-aligned, 16 distinct banks across the
// 16-lane fragment read), B rows 68 dwords (lane halves 8 banks apart).
__global__ __launch_bounds__(128)
void gnncl_gemm(const float* __restrict__ A, const float* __restrict__ B,
                float* __restrict__ D, int K, int lda, int ldb, int ldd) {
  __shared__ float As[2][64][36];   // 64x32 tile + pad  (9216 B per buffer)
  __shared__ float Bs[2][32][68];   // 32x64 tile + pad  (8704 B per buffer)

  const int tid  = threadIdx.x;
  const int lane = tid & 31;
  const int wave = tid >> 5;
  const int r = lane & 15, h = lane >> 4;
  const int mloc = (wave >> 1) * 32;
  const int nloc = (wave & 1) * 32;
  const int m_blk = blockIdx.y * 64;
  const int n_blk = blockIdx.x * 64;

  const unsigned asBase = (unsigned)(uintptr_t)&As[0][0][0];  // LDS offset
  const unsigned bsBase = (unsigned)(uintptr_t)&Bs[0][0][0];

  auto issue = [&](int s, int b) {   // 8 async ops per wave per stage
    const int kt = s * 32;
    #pragma unroll
    for (int i = 0; i < 4; ++i) {
      const int e = i * 128 + tid;
      {  // A tile: 64 rows x 8 b128-quads
        const int row = e >> 3, cq = e & 7;
        const float* g = A + (size_t)(m_blk + row) * lda + kt + cq * 4;
        const unsigned l = asBase + ((unsigned)(b * 64 + row) * 36u + cq * 4u) * 4u;
        asm volatile("global_load_async_to_lds_b128 %0, %1, off"
                     :: "v"(l), "v"(g) : "memory");
      }
      {  // B tile: 32 rows x 16 b128-quads
        const int row = e >> 4, cq = e & 15;
        const float* g = B + (size_t)(kt + row) * ldb + n_blk + cq * 4;
        const unsigned l = bsBase + ((unsigned)(b * 32 + row) * 68u + cq * 4u) * 4u;
        asm volatile("global_load_async_to_lds_b128 %0, %1, off"
                     :: "v"(l), "v"(g) : "memory");
      }
    }
  };

  issue(0, 0);

  v8f acc[2][2] = {};
  const int S = K / 32;
  for (int s = 0; s < S; ++s) {
    const int cur = s & 1;
    if (s + 1 < S) {
      issue(s + 1, cur ^ 1);                             // prefetch next stage
      asm volatile("s_wait_asynccnt 0x8" ::: "memory");  // stage s landed
    } else {
      asm volatile("s_wait_asynccnt 0x0" ::: "memory");
    }
    __syncthreads();                                     // all waves' writes visible
    #pragma unroll
    for (int k = 0; k < 32; k += 4) {
      v2f a[2], b[2];
      #pragma unroll
      for (int f = 0; f < 2; ++f) {
        const float* ap = &As[cur][mloc + f * 16 + r][k + 2 * h];
        a[f].x = ap[0]; a[f].y = ap[1];                  // ds_load_b64
        b[f].x = Bs[cur][k + 2 * h][nloc + f * 16 + r];
        b[f].y = Bs[cur][k + 2 * h + 1][nloc + f * 16 + r];
      }
      #pragma unroll
      for (int fm = 0; fm < 2; ++fm)
        #pragma unroll
        for (int fn = 0; fn < 2; ++fn)
          acc[fm][fn] = __builtin_amdgcn_wmma_f32_16x16x4_f32(
              false, a[fm], false, b[fn], (short)0, acc[fm][fn], false, false);
    }
    __syncthreads();                                     // done reading buf[cur]
  }

  #pragma unroll
  for (int fm = 0; fm < 2; ++fm)
    #pragma unroll
    for (int fn = 0; fn < 2; ++fn)
      #pragma unroll
      for (int v = 0; v < 8; ++v)
        D[(size_t)(m_blk + mloc + fm * 16 + v + 8 * h) * ldd +
          (n_blk + nloc + fn * 16 + r)] = acc[fm][fn][v];
}

// ---------------------------------------------------------------------------
extern "C" void kernel_launch(void* const* d_in, const int* in_sizes, int n_in,
                              void* d_out, int out_size, void* d_ws, size_t ws_size,
                              hipStream_t stream) {
  (void)in_sizes; (void)n_in; (void)out_size; (void)ws_size;

  const float* X  = (const float*)d_in[0];   // [2048,64]
  const float* EA = (const float*)d_in[1];   // [65536]
  const float* GT = (const float*)d_in[2];   // [2048,2048]
  const float* W  = (const float*)d_in[3];   // [64]
  const float* Bb = (const float*)d_in[4];   // [1]
  const int*   EI = (const int*)d_in[5];     // [2,65536]

  // outputs: x_new, A_new, C, loss, Gb (flat, in return order)
  float* out   = (float*)d_out;
  float* x_new = out;                                  // 2048*64
  float* A_new = x_new + (size_t)NN * NF;              // 2048^2
  float* Cout  = A_new + (size_t)NN * NN;              // 2048^2
  float* loss  = Cout  + (size_t)NN * NN;              // 1
  float* Gb    = loss + 1;                             // 2048^2 (pristine G)

  // workspace
  float*  ws   = (float*)d_ws;
  float*  Y    = ws;                                   // 2048*64
  float*  sv   = Y + (size_t)NN * NF;                  // 2048
  float*  G    = sv + NN;                              // 2048^2 (Jacobi working)
  float*  V    = G + (size_t)NN * NN;                  // 2048^2 (eigenvectors)
  float*  T    = V + (size_t)NN * NN;                  // 2048^2 (A@C temp)
  float2* rot  = (float2*)(T + (size_t)NN * NN);       // 1024
  float*  part = (float*)(rot + 1024);                 // 2048 BCE partials
  int*    perm = (int*)(part + 2048);                  // 2048
  float*  sq   = (float*)(perm + 2048);                // 2048

  // 1) prep
  gnncl_prep<<<NN / 8, 256, 0, stream>>>(X, W, Y, sv);
  // 2) distance matrix + sigmoid + BCE partials (WMMA)
  gnncl_dist<<<dim3(16, 128), 256, 0, stream>>>(Y, X, sv, Bb, GT, G, Gb, part);
  // 3) BCE finalize
  gnncl_bce_final<<<1, 1024, 0, stream>>>(part, loss);
  // 4) Jacobi eigensolver: 6 sweeps x 2047 rounds
  gnncl_init_identity<<<(NN * NN) / 256, 256, 0, stream>>>(V);
  for (int sweep = 0; sweep < 6; ++sweep) {
    for (int r = 0; r < NN - 1; ++r) {
      gnncl_jrot<<<1, 1024, 0, stream>>>(G, rot, r);
      gnncl_jcols<<<(NN * 1024) / 256, 256, 0, stream>>>(G, V, rot, r);
      gnncl_jrows<<<(NN * 1024) / 256, 256, 0, stream>>>(G, rot, r);
    }
  }
  // 5) sort eigenvalues ascending, C = V[:,perm] * sqrt(clamp(lambda))
  gnncl_eigsort<<<1, 1024, 0, stream>>>(G, perm, sq);
  gnncl_make_C<<<(NN * NN) / 256, 256, 0, stream>>>(V, perm, sq, Cout);
  // 6) x_new = C @ X   (M=2048, N=64, K=2048)
  gnncl_gemm<<<dim3(1, NN / 64), 128, 0, stream>>>(Cout, X, x_new, NN, NN, NF, NF);
  // 7) adjacency scatter into the A_new region (overwritten by final GEMM)
  gnncl_zero<<<(NN * NN) / 256, 256, 0, stream>>>(A_new);
  gnncl_scatter<<<NE / 256, 256, 0, stream>>>(EI, EA, A_new);
  // 8) T = A @ C ; A_new = C @ T
  gnncl_gemm<<<dim3(NN / 64, NN / 64), 128, 0, stream>>>(A_new, Cout, T, NN, NN, NN, NN);
  gnncl_gemm<<<dim3(NN / 64, NN / 64), 128, 0, stream>>>(Cout, T, A_new, NN, NN, NN, NN);
}